// CLIPVisionTower_Nuwa_abli_7610682049080
// MI455X (gfx1250) — compile-verified
//
#include <hip/hip_runtime.h>
#include <hip/hip_bf16.h>

typedef float v2f __attribute__((ext_vector_type(2)));
typedef float v8f __attribute__((ext_vector_type(8)));

#define B_     8
#define WGRID  24
#define P_     576
#define PP1    577
#define TOPN   3
#define K_     42
#define MPAD   48
#define HEADS_ 16
#define D_     1024
#define NL_    9
#define NCAND  432

// ---------------------------------------------------------------- helpers
__device__ __forceinline__ float dist_pen(int q, int p) {
    float qy = (float)(q / WGRID), qx = (float)(q % WGRID);
    float py = (float)(p / WGRID), px = (float)(p % WGRID);
    float dy = qy - py, dx = qx - px;
    float d  = sqrtf(dy * dy + dx * dx);
    const float inv_th = 0.0597614304667196787f; // 1/sqrt(280)
    return 1.0f - fminf(d * inv_th, 1.0f);
}

// ---------------------------------------------------------------- kernel 1: metric = sum over heads of CLS row
// also zero-fills the shared zero-row used by k_sim's padded A lanes
__global__ void k_metric(const float* __restrict__ attn, float* __restrict__ metric,
                         float* __restrict__ zrow) {
    int idx = blockIdx.x * blockDim.x + threadIdx.x;
    if (idx < D_) zrow[idx] = 0.f;                 // ws is poisoned -> must write
    if (idx >= B_ * P_) return;
    int b = idx / P_, p = idx % P_;
    size_t base = (size_t)b * HEADS_ * PP1 * PP1 + (size_t)(1 + p);
    float s = 0.f;
#pragma unroll
    for (int h = 0; h < HEADS_; ++h)
        s += attn[base + (size_t)h * PP1 * PP1];
    metric[idx] = s;
}

// ---------------------------------------------------------------- kernel 2: region top-3 + global top-K selection
__global__ void k_select(const float* __restrict__ metric, int* __restrict__ bench,
                         float* __restrict__ rowsum, float* __restrict__ out_bench) {
    int b = blockIdx.x;
    int t = threadIdx.x;
    __shared__ float sm[P_];
    __shared__ float cs[NCAND];
    __shared__ int   cg[NCAND];
    __shared__ int   flag[P_];
    for (int i = t; i < P_; i += blockDim.x) { sm[i] = metric[b * P_ + i]; flag[i] = 0; }
    if (t < MPAD) rowsum[b * MPAD + t] = 0.f;      // zero before atomics
    __syncthreads();
    if (t < 144) {
        int ry = t / 12, rx = t % 12;
        float v[4]; int g[4];
#pragma unroll
        for (int loc = 0; loc < 4; ++loc) {
            int ly = loc >> 1, lx = loc & 1;
            int gi = (ry * 2 + ly) * WGRID + (rx * 2 + lx);
            g[loc] = gi; v[loc] = sm[gi];
        }
        bool used[4] = {false, false, false, false};
#pragma unroll
        for (int j = 0; j < TOPN; ++j) {           // top_k(4,3): ties -> lower index first
            int best = 0; float bv = -3.0e38f;
#pragma unroll
            for (int loc = 0; loc < 4; ++loc)
                if (!used[loc] && v[loc] > bv) { bv = v[loc]; best = loc; }
            used[best] = true;
            cs[t * TOPN + j] = bv;
            cg[t * TOPN + j] = g[best];
        }
    }
    __syncthreads();
    if (t < NCAND) {                               // rank = #candidates strictly ahead
        float si = cs[t];
        int rank = 0;
        for (int j = 0; j < NCAND; ++j) {
            float sj = cs[j];
            rank += (sj > si) || (sj == si && j < t);
        }
        if (rank < K_) flag[cg[t]] = 1;            // candidate patches are distinct
    }
    __syncthreads();
    if (t == 0) {                                  // ascending-patch scan == jnp.sort(bench)
        int cnt = 0;
        for (int p = 0; p < P_; ++p)
            if (flag[p]) {
                bench[b * K_ + cnt]     = p;
                out_bench[b * K_ + cnt] = (float)p;
                ++cnt;
            }
    }
}

// ---------------------------------------------------------------- kernel 3: 9-layer mean + row normalize (BW-bound)
__global__ void k_avgnorm(const float* __restrict__ hs, float* __restrict__ nrm) {
    int row = blockIdx.x;                  // 0 .. B*P-1
    int b = row / P_, p = row % P_;
    int t = threadIdx.x;                   // 256 threads, float4 each -> 1024 floats/row
    size_t col = (size_t)t * 4;
    float4 acc = {0.f, 0.f, 0.f, 0.f};
    for (int l = 0; l < NL_; ++l) {
        const float* src = hs + (((size_t)l * B_ + b) * PP1 + (p + 1)) * D_ + col;
        if (l + 1 < NL_)
            __builtin_prefetch(hs + (((size_t)(l + 1) * B_ + b) * PP1 + (p + 1)) * D_ + col, 0, 0);
        float4 v = *(const float4*)src;
        acc.x += v.x; acc.y += v.y; acc.z += v.z; acc.w += v.w;
    }
    const float inv9 = 1.0f / 9.0f;
    acc.x *= inv9; acc.y *= inv9; acc.z *= inv9; acc.w *= inv9;
    float ss = acc.x * acc.x + acc.y * acc.y + acc.z * acc.z + acc.w * acc.w;
    __shared__ float red[256];
    red[t] = ss; __syncthreads();
    for (int s = 128; s > 0; s >>= 1) { if (t < s) red[t] += red[t + s]; __syncthreads(); }
    float sc = 1.0f / fmaxf(sqrtf(red[0]), 1e-12f);
    float4 o = {acc.x * sc, acc.y * sc, acc.z * sc, acc.w * sc};
    *(float4*)(nrm + (size_t)row * D_ + col) = o;
}

// ---------------------------------------------------------------- kernel 4a: sim tiles via V_WMMA_F32_16X16X4_F32
// one wave per 16x16 tile: grid = B * 3(Mtiles) * 36(Ntiles)
// software-pipelined K loop; pad lanes read a shared zero row (no exec masking)
__global__ void k_sim(const float* __restrict__ nrm, const int* __restrict__ bench,
                      const float* __restrict__ zrow,
                      float* __restrict__ wraw, float* __restrict__ rowsum) {
    int wid = blockIdx.x;
    int b   = wid / (3 * 36);
    int rem = wid % (3 * 36);
    int mt = rem / 36, nt = rem % 36;
    int m0 = mt * 16, p0 = nt * 16;
    int lane = threadIdx.x;
    int half = lane >> 4, l16 = lane & 15;

    int  am     = m0 + l16;
    bool avalid = (am < K_);
    int  arow   = avalid ? bench[b * K_ + am] : 0;
    const float* Abase = avalid ? (nrm + ((size_t)b * P_ + arow) * D_ + 2 * half)
                                : (zrow + 2 * half);           // unconditional loads
    const float* Bbase = nrm + ((size_t)b * P_ + (p0 + l16)) * D_ + 2 * half;

    v8f c = {0.f, 0.f, 0.f, 0.f, 0.f, 0.f, 0.f, 0.f};
    v2f a0 = *(const v2f*)(Abase);
    v2f b0 = *(const v2f*)(Bbase);
    for (int d = 0; d < D_ - 4; d += 4) {          // double-buffer: loads fly during wmma
        v2f a1 = *(const v2f*)(Abase + d + 4);
        v2f b1 = *(const v2f*)(Bbase + d + 4);
        c = __builtin_amdgcn_wmma_f32_16x16x4_f32(false, a0, false, b0, (short)0, c, false, false);
        a0 = a1; b0 = b1;
    }
    c = __builtin_amdgcn_wmma_f32_16x16x4_f32(false, a0, false, b0, (short)0, c, false, false);

    int p = p0 + l16;
#pragma unroll
    for (int r = 0; r < 8; ++r) {
        int m = m0 + r + 8 * half;
        int q = (m < K_) ? bench[b * K_ + m] : 0;
        float v = fmaxf(c[r], 0.f) * dist_pen(q, p);
        if (m >= K_) v = 0.f;                      // pad rows
        wraw[((size_t)b * MPAD + m) * P_ + p] = v;
        float s = v;                               // reduce across the 16 lanes of this half
        s += __shfl_xor(s, 1, 32);
        s += __shfl_xor(s, 2, 32);
        s += __shfl_xor(s, 4, 32);
        s += __shfl_xor(s, 8, 32);
        if (l16 == 0) atomicAdd(&rowsum[b * MPAD + m], s);
    }
}

// ---------------------------------------------------------------- kernel 4b: out = w @ patch_agg via WMMA f32
// one wave per 16x16 tile: grid = B * 3(Mtiles) * 64(Ntiles over D), pipelined K loop
__global__ void k_out(const float* __restrict__ wraw, const float* __restrict__ rowsum,
                      const int* __restrict__ bench, const float* __restrict__ hidden,
                      float* __restrict__ out) {
    int wid = blockIdx.x;
    int b   = wid / (3 * 64);
    int rem = wid % (3 * 64);
    int mt = rem / 64, nt = rem % 64;
    int m0 = mt * 16, n0 = nt * 16;
    int lane = threadIdx.x;
    int half = lane >> 4, l16 = lane & 15;

    int   m   = m0 + l16;                          // A-row owned by this lane
    float inv = 1.0f / (rowsum[b * MPAD + m] + 1e-8f);
    int   bm  = (m < K_) ? bench[b * K_ + m] : -1;
    const float* Abase = wraw + ((size_t)b * MPAD + m) * P_ + 2 * half;
    const float* Bcol  = hidden + (size_t)b * PP1 * D_ + (size_t)(n0 + l16);

    auto loadA = [&](int pk) -> v2f {
        v2f raw = *(const v2f*)(Abase + pk);
        int pA = pk + 2 * half;
        v2f a;
        a.x = (pA     == bm) ? 1.0f : raw.x * inv; // diag override AFTER normalization
        a.y = (pA + 1 == bm) ? 1.0f : raw.y * inv;
        return a;
    };
    auto loadB = [&](int pk) -> v2f {
        int pB = pk + 2 * half;
        v2f bb;
        bb.x = Bcol[(size_t)(pB + 1) * D_];        // +1 skips CLS row
        bb.y = Bcol[(size_t)(pB + 2) * D_];
        return bb;
    };

    v8f c = {0.f, 0.f, 0.f, 0.f, 0.f, 0.f, 0.f, 0.f};
    v2f a0 = loadA(0);
    v2f b0 = loadB(0);
    for (int pk = 0; pk < P_ - 4; pk += 4) {       // double-buffered
        v2f a1 = loadA(pk + 4);
        v2f b1 = loadB(pk + 4);
        c = __builtin_amdgcn_wmma_f32_16x16x4_f32(false, a0, false, b0, (short)0, c, false, false);
        a0 = a1; b0 = b1;
    }
    c = __builtin_amdgcn_wmma_f32_16x16x4_f32(false, a0, false, b0, (short)0, c, false, false);

#pragma unroll
    for (int r = 0; r < 8; ++r) {
        int mm = m0 + r + 8 * half;
        if (mm < K_)
            out[((size_t)b * K_ + mm) * D_ + n0 + l16] = c[r];
    }
}

// ---------------------------------------------------------------- launch
extern "C" void kernel_launch(void* const* d_in, const int* in_sizes, int n_in,
                              void* d_out, int out_size, void* d_ws, size_t ws_size,
                              hipStream_t stream) {
    (void)in_sizes; (void)n_in; (void)out_size; (void)ws_size;
    const float* attn   = (const float*)d_in[0];   // (8,16,577,577) f32
    const float* hidden = (const float*)d_in[1];   // (8,577,1024)   f32
    const float* hs     = (const float*)d_in[2];   // (9,8,577,1024) f32
    float* out = (float*)d_out;

    // ws layout (float elements): metric | bench(int) | rowsum | wraw | nrm | zrow
    float* wsf    = (float*)d_ws;
    float* metric = wsf;                           // 8*576      = 4608
    int*   bench  = (int*)(wsf + 4608);            // 8*42       = 336
    float* rowsum = wsf + 4944;                    // 8*48       = 384
    float* wraw   = wsf + 5328;                    // 8*48*576   = 221184
    float* nrm    = wsf + 226512;                  // 8*576*1024 = 4718592
    float* zrow   = wsf + 4945104;                 // 1024 zeros (~18.9 MB total)
    float* out_bench = out + (size_t)B_ * K_ * D_;

    k_metric <<<(B_ * P_ + 255) / 256, 256, 0, stream>>>(attn, metric, zrow);
    k_select <<<B_, 512, 0, stream>>>(metric, bench, rowsum, out_bench);
    k_avgnorm<<<B_ * P_, 256, 0, stream>>>(hs, nrm);
    k_sim    <<<B_ * 3 * 36, 32, 0, stream>>>(nrm, bench, zrow, wraw, rowsum);
    k_out    <<<B_ * 3 * 64, 32, 0, stream>>>(wraw, rowsum, bench, hidden, out);
}